// BVHGateWrapper_65137474011768
// MI455X (gfx1250) — compile-verified
//
#include <hip/hip_runtime.h>
#include <hip/hip_bf16.h>

// MoE router: fp32 GEMM [T,2048]x[2048,64] via v_wmma_f32_16x16x4_f32,
// async-to-LDS weight staging (ASYNCcnt path), fused affine calibration +
// fp32 softmax + stable top-8.

typedef __attribute__((ext_vector_type(2))) float v2f;
typedef __attribute__((ext_vector_type(4))) float v4f;
typedef __attribute__((ext_vector_type(8))) float v8f;
typedef __attribute__((ext_vector_type(4))) int   v4i;

typedef __attribute__((address_space(1))) v4i* glb_v4i_ptr;  // global
typedef __attribute__((address_space(3))) v4i* lds_v4i_ptr;  // LDS

#define TOKENS 16384
#define DHID   2048
#define NEXP   64
#define TOPK   8

#define KC        64                 // K-chunk staged in LDS
#define PAD       4                  // dword padding -> conflict-free ds_load_b64
#define ROWSTRIDE (KC + PAD)         // 68 floats, 272B (16B aligned rows)
#define NCHUNK    (DHID / KC)        // 32
#define WAVES_PER_BLOCK 8
#define TOK_PER_WAVE    16
#define TOK_PER_BLOCK   (WAVES_PER_BLOCK * TOK_PER_WAVE)  // 128

#if defined(__has_builtin)
#if __has_builtin(__builtin_amdgcn_global_load_async_to_lds_b128)
#define HAVE_ASYNC_LDS 1
#endif
#endif

__device__ __forceinline__ void wait_asynccnt0() {
#if defined(HAVE_ASYNC_LDS)
#if __has_builtin(__builtin_amdgcn_s_wait_asynccnt)
    __builtin_amdgcn_s_wait_asynccnt(0);
#else
    asm volatile("s_wait_asynccnt 0x0" ::: "memory");
#endif
#endif
}

__global__ __launch_bounds__(256, 1)
void moe_router_wmma(const float* __restrict__ h,      // [T, D]
                     const float* __restrict__ w,      // [E, D]
                     const float* __restrict__ scale,  // [E]
                     const float* __restrict__ bias,   // [E]
                     float* __restrict__ probs,        // [T, E]
                     float* __restrict__ topw,         // [T, K]
                     int*   __restrict__ topi)         // [T, K]
{
    // 2 weight buffers (2*64*68 = 8704 floats = 34.8KB); phase-2 logits
    // (8 waves * 16 tok * 64 exp = 8192 floats) alias the same storage.
    __shared__ __align__(16) float lds[2 * NEXP * ROWSTRIDE];

    const int tid  = threadIdx.x;
    const int wave = tid >> 5;
    const int lane = tid & 31;
    const int row  = lane & 15;        // A: token-in-tile (M); B: expert (N) low bits
    const int kp   = (lane >> 4) * 2;  // K sub-offset inside a K=4 step

    const int tokenBase = blockIdx.x * TOK_PER_BLOCK + wave * TOK_PER_WAVE;
    const float* aPtr = h + (size_t)(tokenBase + row) * DHID + kp;

    v8f acc0 = {}, acc1 = {}, acc2 = {}, acc3 = {};

    // Cooperative stage of W[0:64][k0:k0+KC] into LDS buffer `buf`.
    // gfx1250 async-to-LDS path (cache -> LDS, no VGPR roundtrip, ASYNCcnt);
    // falls back to load+ds_store if the builtin is absent.
    auto stage = [&](int chunk, int buf) {
        const int k0 = chunk * KC;
        float* dst = &lds[buf * NEXP * ROWSTRIDE];
        #pragma unroll
        for (int i = 0; i < 4; ++i) {
            int id = tid + i * 256;      // 1024 float4 tiles total
            int e  = id >> 4;            // KC/4 = 16 float4 per expert row
            int j4 = id & 15;
            const float* src = w + (size_t)e * DHID + k0 + j4 * 4;
            float*       d   = dst + e * ROWSTRIDE + j4 * 4;
#if defined(HAVE_ASYNC_LDS)
            __builtin_amdgcn_global_load_async_to_lds_b128(
                (glb_v4i_ptr)src, (lds_v4i_ptr)d,
                /*imm offset=*/0, /*cpol=*/0);
#else
            *(v4f*)d = *(const v4f*)src;
#endif
        }
    };

    stage(0, 0);
    wait_asynccnt0();
    __syncthreads();

    for (int c = 0; c < NCHUNK; ++c) {
        const int buf = c & 1;
        if (c + 1 < NCHUNK) stage(c + 1, buf ^ 1);  // double-buffer next chunk

        const float* a = aPtr + c * KC;
        __builtin_prefetch(a + KC, 0, 0);           // global_prefetch next A chunk
        const float* bBase = &lds[buf * NEXP * ROWSTRIDE];

        #pragma unroll
        for (int s = 0; s < KC / 4; ++s) {
            // A 16x4 fp32 fragment: VGPR0=K(kp), VGPR1=K(kp+1) per ISA layout
            v2f av = *(const v2f*)(a + 4 * s);
            // B 4x16 fp32 fragments for the 4 expert tiles (bank-conflict-free)
            v2f b0 = *(const v2f*)(bBase + (row +  0) * ROWSTRIDE + 4 * s + kp);
            v2f b1 = *(const v2f*)(bBase + (row + 16) * ROWSTRIDE + 4 * s + kp);
            v2f b2 = *(const v2f*)(bBase + (row + 32) * ROWSTRIDE + 4 * s + kp);
            v2f b3 = *(const v2f*)(bBase + (row + 48) * ROWSTRIDE + 4 * s + kp);
            acc0 = __builtin_amdgcn_wmma_f32_16x16x4_f32(false, av, false, b0, (short)0, acc0, false, false);
            acc1 = __builtin_amdgcn_wmma_f32_16x16x4_f32(false, av, false, b1, (short)0, acc1, false, false);
            acc2 = __builtin_amdgcn_wmma_f32_16x16x4_f32(false, av, false, b2, (short)0, acc2, false, false);
            acc3 = __builtin_amdgcn_wmma_f32_16x16x4_f32(false, av, false, b3, (short)0, acc3, false, false);
        }
        wait_asynccnt0();  // this wave's async stage of buf^1 is complete
        __syncthreads();   // all waves done reading buf; stage writes visible
    }

    // ---- epilogue: calibrate, dump logits to LDS (alias weight buffers) ----
    float* wl = &lds[wave * TOK_PER_WAVE * NEXP];   // 8*1024 <= 8704 floats
    {
        const int e0 = row;
        const float s0 = scale[e0],      b0 = bias[e0];
        const float s1 = scale[e0 + 16], b1 = bias[e0 + 16];
        const float s2 = scale[e0 + 32], b2 = bias[e0 + 32];
        const float s3 = scale[e0 + 48], b3 = bias[e0 + 48];
        const int thalf = (lane >> 4) * 8;   // C/D layout: VGPR r -> M = r + 8*(lane>>4)
        #pragma unroll
        for (int r = 0; r < 8; ++r) {
            const int t = r + thalf;
            wl[t * NEXP + e0     ] = acc0[r] * s0 + b0;
            wl[t * NEXP + e0 + 16] = acc1[r] * s1 + b1;
            wl[t * NEXP + e0 + 32] = acc2[r] * s2 + b2;
            wl[t * NEXP + e0 + 48] = acc3[r] * s3 + b3;
        }
    }
    __syncthreads();

    // ---- softmax (fp32) + stable top-8, one lane per token ----
    if (lane < 16) {
        const int t = lane;
        const float* lg = wl + t * NEXP;

        float m = lg[0];
        for (int e = 1; e < NEXP; ++e) m = fmaxf(m, lg[e]);

        // register-resident descending top-8 (stable: ties keep lower index)
        float tv[TOPK]; int ti[TOPK];
        #pragma unroll
        for (int j = 0; j < TOPK; ++j) { tv[j] = -3.402823466e38f; ti[j] = 0; }

        float sum = 0.f;
        for (int e = 0; e < NEXP; ++e) {
            const float val = lg[e];
            sum += __expf(val - m);
            float v = val; int ie = e;
            #pragma unroll
            for (int j = 0; j < TOPK; ++j) {
                const bool  gt = v > tv[j];
                const float ov = tv[j]; const int oi = ti[j];
                tv[j] = gt ? v  : ov;
                ti[j] = gt ? ie : oi;
                v     = gt ? ov : v;
                ie    = gt ? oi : ie;
            }
        }
        const float inv = 1.f / sum;

        const int gt = tokenBase + t;
        float* pout = probs + (size_t)gt * NEXP;
        for (int e = 0; e < NEXP; ++e) pout[e] = __expf(lg[e] - m) * inv;

        #pragma unroll
        for (int j = 0; j < TOPK; ++j) {
            topw[(size_t)gt * TOPK + j] = __expf(tv[j] - m) * inv;
            topi[(size_t)gt * TOPK + j] = ti[j];
        }
    }
}

extern "C" void kernel_launch(void* const* d_in, const int* in_sizes, int n_in,
                              void* d_out, int out_size, void* d_ws, size_t ws_size,
                              hipStream_t stream) {
    const float* h     = (const float*)d_in[0];   // hidden_states [4,4096,2048] f32
    const float* w     = (const float*)d_in[1];   // router_weight [64,2048] f32
    const float* scale = (const float*)d_in[2];   // [64] f32
    const float* bias  = (const float*)d_in[3];   // [64] f32

    float* probs = (float*)d_out;                          // [T,64] f32
    float* topw  = probs + (size_t)TOKENS * NEXP;          // [T,8]  f32
    int*   topi  = (int*)(topw + (size_t)TOKENS * TOPK);   // [T,8]  i32 (raw bits)

    dim3 grid(TOKENS / TOK_PER_BLOCK);  // 128 blocks x 256 threads (8 waves)
    moe_router_wmma<<<grid, 256, 0, stream>>>(h, w, scale, bias, probs, topw, topi);
}